// ConvCaps_5832565588545
// MI455X (gfx1250) — compile-verified
//
#include <hip/hip_runtime.h>
#include <hip/hip_bf16.h>

// ---------------------------------------------------------------------------
// ConvCaps EM-routing forward for MI455X (gfx1250, wave32).
//  - votes einsum lowered to V_WMMA_F32_16X16X4_F32 (K=4 == H, exact fit)
//  - votes tensor (170MB fp32) parked in global memory; fits MI455X 192MB L2
//  - EM routing: memory-bound; per-(cw,h) transcendentals (log/sqrt/rcp of
//    sigma stats) hoisted out of the Bkk-redundant R-update into em_stats
// ---------------------------------------------------------------------------

#define B_CAPS   32
#define C_CAPS   32
#define KSZ      3
#define HSZ      4
#define HH       16
#define WIN      14
#define WOUT     6
#define WW       36          // WOUT*WOUT
#define BKK      288         // B_CAPS*KSZ*KSZ
#define CWW      1152        // C_CAPS*WW
#define NB       8           // batch
#define EPS      1e-10f
#define HALF_LN2PI 0.9189385332046727f

typedef __attribute__((ext_vector_type(2))) float v2f;
typedef __attribute__((ext_vector_type(8))) float v8f;

// ---------------------------------------------------------------------------
// Kernel 1: votes[n,eij,c*36+s,x*4+z] = sum_y W[eij,c,x,y] * P[n,eij,s,y,z]
// One block per (n,eij) group (2304 blocks). 8 waves/block, one M-tile each.
// Per wave: load A (16x4 slice of W, 2 VGPRs) once, loop 9 N-tiles of B,
// issue v_wmma_f32_16x16x4_f32, scatter-store the 16x16 f32 D tile.
// ---------------------------------------------------------------------------
__global__ __launch_bounds__(256) void votes_wmma_kernel(
    const float* __restrict__ poses,   // (8,32,14,14,16)
    const float* __restrict__ W,       // (32,3,3,32,4,4)
    float* __restrict__ V)             // (8,288,1152,16)
{
  const int group = blockIdx.x;                 // n*288 + eij
  const int n   = group / BKK;
  const int eij = group % BKK;
  const int i   = (eij / 3) % 3;
  const int j   = eij % 3;

  const int mt   = threadIdx.x >> 5;            // wave id == M tile (0..7)
  const int lane = threadIdx.x & 31;
  const int l16  = lane & 15;
  const int half = lane >> 4;                   // 0: K=0,1  1: K=2,3
  const int ybase = half * 2;

  // ---- A matrix: rows M = mt*16 + l16 ; M = c*4 + x ----
  const int c_m = mt * 4 + (l16 >> 2);
  const int x_m = l16 & 3;
  const float* Wp = W + ((size_t)(eij * C_CAPS + c_m)) * HH + x_m * HSZ + ybase;
  v2f amat; amat.x = Wp[0]; amat.y = Wp[1];     // y and y+1 contiguous

  const float* Pn   = poses + (size_t)(n * B_CAPS + (eij / 9)) * WIN * WIN * HH;
  float* __restrict__ Vout = V + (size_t)group * CWW * HH;

  for (int nt = 0; nt < 9; ++nt) {
    // ---- B matrix: col N = nt*16 + l16 ; N = s*4 + z, s = v*6 + w ----
    const int Ng  = nt * 16 + l16;
    const int s   = Ng >> 2;
    const int z   = Ng & 3;
    const int vv  = s / WOUT;                   // slow spatial index
    const int wwi = s % WOUT;                   // fast spatial index
    const int prow = 2 * wwi + i;               // pose row uses fast index
    const int pcol = 2 * vv + j;                // pose col uses slow index
    const float* Pp = Pn + ((size_t)(prow * WIN + pcol)) * HH + z;
    v2f bmat;
    bmat.x = Pp[ybase * HSZ];                   // y = ybase
    bmat.y = Pp[(ybase + 1) * HSZ];             // y = ybase+1

    v8f acc = {};
    acc = __builtin_amdgcn_wmma_f32_16x16x4_f32(
        /*neg_a=*/false, amat, /*neg_b=*/false, bmat,
        /*c_mod=*/(short)0, acc, /*reuse_a=*/false, /*reuse_b=*/false);

    // ---- D tile: VGPR r, lane: M = mt*16 + r + 8*half, N = l16 ----
#pragma unroll
    for (int r = 0; r < 8; ++r) {
      const int M  = mt * 16 + r + 8 * half;
      const int cc = M >> 2;
      const int xx = M & 3;
      Vout[((size_t)(cc * WW + s)) * HH + xx * HSZ + z] = acc[r];
    }
  }
}

// ---------------------------------------------------------------------------
// Kernel 2: EM statistics. One block per (n, cw): reduce over Bkk=288.
// 256 threads: lane-h = t%16 (coalesced 64B V reads), k-group = t/16 (x18).
// sigma_sq via E[V^2] - mu^2 (single pass over V). Emits the derived
// quantities the later kernels need, so log/sqrt/rcp happen once per (cw,h):
//   lsc  = log(sqrt(sigma_sq)+EPS) + 0.5*ln(2*pi)
//   inv2 = 1/(2*sigma_sq)
// ---------------------------------------------------------------------------
template <bool FIRST>
__global__ __launch_bounds__(256) void em_stats_kernel(
    const float* __restrict__ V,      // (8,288,1152,16)
    const float* __restrict__ R,      // (8,288,1152)   (unused when FIRST)
    const float* __restrict__ acts,   // (8,32,14,14)
    float* __restrict__ sumR,         // (8,1152)
    float* __restrict__ mu,           // (8,1152,16)
    float* __restrict__ lsc,          // (8,1152,16)
    float* __restrict__ inv2)         // (8,1152,16)
{
  const int bcw = blockIdx.x;                   // n*1152 + cw
  const int n   = bcw / CWW;
  const int cw  = bcw % CWW;
  const int s   = cw % WW;
  const int vv  = s / WOUT;
  const int wwi = s % WOUT;

  const int t    = threadIdx.x;
  const int h    = t & 15;
  const int kgrp = t >> 4;                      // 0..15, each covers 18 k

  float acc_r = 0.f, acc_mu = 0.f, acc_sg = 0.f;
  for (int kk = 0; kk < 18; ++kk) {
    const int k = kgrp * 18 + kk;
    const int e = k / 9;
    const int i = (k / 3) % 3;
    const int j = k % 3;
    const float aval =
        acts[((n * B_CAPS + e) * WIN + (2 * wwi + i)) * WIN + (2 * vv + j)];
    const float r =
        FIRST ? (1.0f / (float)C_CAPS)
              : R[((size_t)(n * BKK + k)) * CWW + cw];
    const float rw = r * aval;
    const size_t vidx = (((size_t)(n * BKK + k)) * CWW + cw) * HH + h;
    const float v = V[vidx];
    if (kk + 1 < 18) {
      // k-stride is ~73KB: hint the next line (global_prefetch_b8)
      __builtin_prefetch(&V[vidx + (size_t)CWW * HH], 0, 1);
    }
    acc_r  += rw;
    acc_mu += rw * v;
    acc_sg += rw * v * v;
  }

  __shared__ float s_mu[16][16];
  __shared__ float s_sg[16][16];
  __shared__ float s_r[16];
  s_mu[kgrp][h] = acc_mu;
  s_sg[kgrp][h] = acc_sg;
  if (h == 0) s_r[kgrp] = acc_r;                // rw is h-independent
  __syncthreads();

  if (t < 16) {
    float mnum = 0.f, snum = 0.f, rsum = 0.f;
#pragma unroll
    for (int g = 0; g < 16; ++g) {
      mnum += s_mu[g][t];
      snum += s_sg[g][t];
      rsum += s_r[g];
    }
    const float mu_v = mnum / rsum;
    float sg_v = snum / rsum - mu_v * mu_v;
    sg_v = fmaxf(sg_v, 1e-30f);
    const float ls = logf(sqrtf(sg_v) + EPS);
    mu[(size_t)bcw * HH + t]   = mu_v;
    lsc[(size_t)bcw * HH + t]  = ls + HALF_LN2PI;
    inv2[(size_t)bcw * HH + t] = 0.5f / sg_v;
    if (t == 0) sumR[bcw] = rsum;
  }
}

// ---------------------------------------------------------------------------
// Kernel 3: activation. a = sigmoid(lambda*(beta_a[c] - sum_h (beta_v[c]+ls)*sumR))
// ls recovered from lsc by folding -0.5*ln(2*pi) into beta_v once.
// ---------------------------------------------------------------------------
__global__ __launch_bounds__(256) void em_act_kernel(
    const float* __restrict__ lsc,     // (8,1152,16)
    const float* __restrict__ sumR,    // (8,1152)
    const float* __restrict__ beta_v,  // (32)
    const float* __restrict__ beta_a,  // (32)
    const float* __restrict__ lambda_p,
    float* __restrict__ a_out)         // (8,1152)
{
  const int idx = blockIdx.x * blockDim.x + threadIdx.x;
  if (idx >= NB * CWW) return;
  const int c  = (idx % CWW) / WW;
  const float sr  = sumR[idx];
  const float bvp = beta_v[c] - HALF_LN2PI;     // beta_v + (lsc - 0.5*ln2pi)
  float cost = 0.f;
#pragma unroll
  for (int h = 0; h < HH; ++h)
    cost += (bvp + lsc[(size_t)idx * HH + h]) * sr;
  const float x = lambda_p[0] * (beta_a[c] - cost);
  a_out[idx] = 1.f / (1.f + expf(-x));
}

// ---------------------------------------------------------------------------
// Kernel 4: R update. One block per (n,k) row; ap[1152] staged in LDS,
// normalized over Cww in-block (wave32 shfl + cross-wave LDS reduce).
// Inner loop: one FMA + one exp per element — all sigma-derived
// transcendentals were precomputed in em_stats.
// ---------------------------------------------------------------------------
__global__ __launch_bounds__(288) void em_update_R_kernel(
    const float* __restrict__ V,       // (8,288,1152,16)
    const float* __restrict__ mu,      // (8,1152,16)
    const float* __restrict__ lsc,     // (8,1152,16)
    const float* __restrict__ inv2,    // (8,1152,16)
    const float* __restrict__ a,       // (8,1152)
    float* __restrict__ R)             // (8,288,1152)
{
  const int blk = blockIdx.x;                   // n*288 + k
  const int n   = blk / BKK;
  const size_t base = (size_t)blk * CWW;

  __shared__ float s_ap[CWW];
  __shared__ float s_red[9];

  float asum = 0.f;
#pragma unroll
  for (int ii = 0; ii < 4; ++ii) {
    const int cw = ii * 288 + threadIdx.x;
    const float* vp = V + (base + cw) * HH;
    const float* mp = mu + ((size_t)n * CWW + cw) * HH;
    const float* lp = lsc + ((size_t)n * CWW + cw) * HH;
    const float* ip = inv2 + ((size_t)n * CWW + cw) * HH;
    float p = 0.f;
#pragma unroll
    for (int h = 0; h < HH; ++h) {
      const float d = vp[h] - mp[h];
      // ln_p = -d^2/(2*sigma^2) - log_sigma - 0.5*ln(2*pi)
      p += __expf(-(d * d) * ip[h] - lp[h]);
    }
    const float ap = a[(size_t)n * CWW + cw] * p;
    s_ap[cw] = ap;
    asum += ap;
  }

  // wave32 reduction, then cross-wave (9 waves)
#pragma unroll
  for (int off = 16; off > 0; off >>= 1)
    asum += __shfl_xor(asum, off, 32);
  if ((threadIdx.x & 31) == 0) s_red[threadIdx.x >> 5] = asum;
  __syncthreads();
  if (threadIdx.x == 0) {
    float tot = 0.f;
#pragma unroll
    for (int g = 0; g < 9; ++g) tot += s_red[g];
    s_red[0] = tot;
  }
  __syncthreads();
  const float tot = s_red[0];
#pragma unroll
  for (int ii = 0; ii < 4; ++ii) {
    const int cw = ii * 288 + threadIdx.x;
    R[base + cw] = s_ap[cw] / (tot + EPS) + EPS;
  }
}

// ---------------------------------------------------------------------------
// Host launcher
// ---------------------------------------------------------------------------
extern "C" void kernel_launch(void* const* d_in, const int* in_sizes, int n_in,
                              void* d_out, int out_size, void* d_ws, size_t ws_size,
                              hipStream_t stream) {
  const float* lambda_p = (const float*)d_in[0];   // scalar on device
  const float* poses    = (const float*)d_in[1];   // (8,32,14,14,16)
  const float* acts     = (const float*)d_in[2];   // (8,32,14,14)
  const float* W        = (const float*)d_in[3];   // (32,3,3,32,4,4)
  const float* beta_v   = (const float*)d_in[4];   // (1,32,1,1)
  const float* beta_a   = (const float*)d_in[5];   // (1,32,1)

  // scratch layout (fp32)
  float* ws    = (float*)d_ws;
  float* V     = ws;                                          // 42,467,328
  float* R     = V + (size_t)NB * BKK * CWW * HH;             //  2,654,208
  float* sumR  = R + (size_t)NB * BKK * CWW;                  //      9,216
  float* mu    = sumR + (size_t)NB * CWW;                     //    147,456
  float* lsc   = mu + (size_t)NB * CWW * HH;                  //    147,456
  float* inv2  = lsc + (size_t)NB * CWW * HH;                 //    147,456
  float* aBuf  = inv2 + (size_t)NB * CWW * HH;                //      9,216

  // 1) votes via WMMA f32 16x16x4 — one block per (n, eij)
  votes_wmma_kernel<<<NB * BKK, 256, 0, stream>>>(poses, W, V);

  // 2) EM routing, 3 iterations
  for (int it = 0; it < 3; ++it) {
    if (it == 0)
      em_stats_kernel<true><<<NB * CWW, 256, 0, stream>>>(V, R, acts, sumR, mu,
                                                          lsc, inv2);
    else
      em_stats_kernel<false><<<NB * CWW, 256, 0, stream>>>(V, R, acts, sumR, mu,
                                                           lsc, inv2);

    em_act_kernel<<<(NB * CWW + 255) / 256, 256, 0, stream>>>(
        lsc, sumR, beta_v, beta_a, lambda_p, aBuf);

    if (it < 2)
      em_update_R_kernel<<<NB * BKK, 288, 0, stream>>>(V, mu, lsc, inv2, aBuf, R);
  }

  // 3) outputs: mu (b,C,w,w,hh) then a (b,C,w,w), both already in flat order
  hipMemcpyAsync(d_out, mu, sizeof(float) * (size_t)NB * CWW * HH,
                 hipMemcpyDeviceToDevice, stream);
  hipMemcpyAsync((float*)d_out + (size_t)NB * CWW * HH, aBuf,
                 sizeof(float) * (size_t)NB * CWW,
                 hipMemcpyDeviceToDevice, stream);
}